// DFEM_36421322670582
// MI455X (gfx1250) — compile-verified
//
#include <hip/hip_runtime.h>

typedef float v2f __attribute__((ext_vector_type(2)));
typedef float v8f __attribute__((ext_vector_type(8)));

// ---------------- workspace layout (float offsets) ----------------
constexpr size_t SZ_WT   = (size_t)9 * 256 * 256;      // 589824 per branch
constexpr size_t OFF_WT_X = 0;
constexpr size_t OFF_WT_Y = OFF_WT_X + SZ_WT;
constexpr size_t OFF_WT_O = OFF_WT_Y + SZ_WT;          // 512*256
constexpr size_t OFF_PWT  = OFF_WT_O + (size_t)512 * 256;   // 256*36*9
constexpr size_t OFF_OFFS = OFF_PWT + (size_t)256 * 36 * 9; // 2 * 4*18*4096
constexpr size_t OFF_CONV = OFF_OFFS + (size_t)2 * 4 * 18 * 4096; // 16384*512

__device__ __forceinline__ v8f wmma4(v2f a, v2f b, v8f c) {
  return __builtin_amdgcn_wmma_f32_16x16x4_f32(false, a, false, b, (short)0, c,
                                               false, false);
}

// ---------------- weight transposes ----------------
__global__ void k_transpose_main_w(const float* __restrict__ w,
                                   float* __restrict__ wT) {
  int i = blockIdx.x * blockDim.x + threadIdx.x;
  if (i >= 9 * 256 * 256) return;
  int t = i >> 16;
  int rem = i & 65535;
  int c = rem >> 8;
  int n = rem & 255;
  wT[i] = w[n * 2304 + c * 9 + t];
}

__global__ void k_transpose_wo(const float* __restrict__ wo,
                               float* __restrict__ wT) {
  int i = blockIdx.x * blockDim.x + threadIdx.x;
  if (i >= 512 * 256) return;
  int k = i >> 8;
  int n = i & 255;
  wT[i] = wo[n * 512 + k];
}

__global__ void k_build_pwt(const float* __restrict__ pwx,
                            const float* __restrict__ pwy,
                            float* __restrict__ pwT) {
  int i = blockIdx.x * blockDim.x + threadIdx.x;
  if (i >= 256 * 36 * 9) return;
  int c = i / 324;
  int r = i - c * 324;
  int oc = r / 9;
  int t = r - oc * 9;
  pwT[i] = (oc < 18) ? pwx[oc * 2304 + c * 9 + t]
                     : pwy[(oc - 18) * 2304 + c * 9 + t];
}

// ---------------- offset conv: 256 -> 36 (both branches), 3x3 pad 1 --------
__global__ __launch_bounds__(256) void k_offset_conv(
    const float* __restrict__ x, const float* __restrict__ pwT,
    const float* __restrict__ pbx, const float* __restrict__ pby,
    float* __restrict__ offs) {
  __shared__ float xs[3 * 66];
  __shared__ float wl[324];
  int tid = threadIdx.x;
  int blk = blockIdx.x;  // 0..255 : (b,h)
  int b = blk >> 6, h = blk & 63;
  int w = tid & 63, g = tid >> 6;  // g in 0..3 -> oc group of 9
  float acc[9];
#pragma unroll
  for (int j = 0; j < 9; ++j) acc[j] = 0.f;

  for (int c = 0; c < 256; ++c) {
    __syncthreads();
    if (tid < 198) {
      int r = tid / 66, col = tid - r * 66;
      int hr = h - 1 + r, wc = col - 1;
      float v = 0.f;
      if (hr >= 0 && hr < 64 && wc >= 0 && wc < 64)
        v = x[((size_t)(b * 256 + c) * 64 + hr) * 64 + wc];
      xs[tid] = v;
    }
    for (int i2 = tid; i2 < 324; i2 += 256) wl[i2] = pwT[c * 324 + i2];
    __syncthreads();

    float xv[9];
#pragma unroll
    for (int r = 0; r < 3; ++r)
#pragma unroll
      for (int kc = 0; kc < 3; ++kc) xv[r * 3 + kc] = xs[r * 66 + w + kc];
#pragma unroll
    for (int j = 0; j < 9; ++j) {
      const float* wp = &wl[(g * 9 + j) * 9];
      float a = acc[j];
#pragma unroll
      for (int t = 0; t < 9; ++t) a = fmaf(xv[t], wp[t], a);
      acc[j] = a;
    }
  }
#pragma unroll
  for (int j = 0; j < 9; ++j) {
    int oc = g * 9 + j;
    int branch = (oc >= 18) ? 1 : 0;
    int ocb = branch ? oc - 18 : oc;
    float bias = branch ? pby[ocb] : pbx[ocb];
    offs[(size_t)branch * (4 * 18 * 4096) + (size_t)(b * 18 + ocb) * 4096 +
         h * 64 + w] = acc[j] + bias;
  }
}

// ---------------- fused deformable-sample + implicit GEMM (WMMA f32) ------
// Block: 32 pixels (M) x 256 out-channels (N), 8 waves, each wave 2x2 tiles.
// K = 9 taps * 256 channels, processed as tap-outer, 32-channel chunks.
__global__ __launch_bounds__(256) void k_deform_gemm(
    const float* __restrict__ x,     // (4,256,64,64)
    const float* __restrict__ offs,  // this branch: (4,18,64,64)
    const float* __restrict__ wT,    // (9,256,256)  [t][c][n]
    float* __restrict__ convout,     // (16384,512)
    int colOff) {
  // pair-interleaved LDS: element q holds K={2q,2q+1}; row stride padded so
  // lanes 0-15 (kpair q) and 16-31 (kpair q+1) hit disjoint bank halves.
  __shared__ v2f As2[16 * 48];   // 32 pixels per row
  __shared__ v2f Bs2[16 * 272];  // 256 N per row
  __shared__ int sidx[4][32];
  __shared__ float swgt[4][32];

  int tid = threadIdx.x;
  int lane = tid & 31;
  int wv = tid >> 5;
  int m0 = blockIdx.x << 5;
  int bT = m0 >> 12;  // batch of this tile (tiles never cross batches)
  int n0 = wv << 5;

  float* AsF = (float*)As2;
  float* BsF = (float*)Bs2;

  v8f acc[2][2];
#pragma unroll
  for (int mi = 0; mi < 2; ++mi)
#pragma unroll
    for (int ni = 0; ni < 2; ++ni)
#pragma unroll
      for (int r = 0; r < 8; ++r) acc[mi][ni][r] = 0.f;

  int sp = tid & 31;          // sampling: pixel
  int skk = tid >> 5;         // sampling: channel base (step 8)
  int bkk = tid >> 3;         // B stage: chunk row 0..31
  int bnb = (tid & 7) << 5;   // B stage: n base

  for (int t = 0; t < 9; ++t) {
    __syncthreads();
    if (tid < 32) {
      int p = tid;
      int hw = (m0 + p) & 4095;
      int h = hw >> 6, w = hw & 63;
      float ox = offs[(size_t)(bT * 18 + t) * 4096 + hw];
      float oy = offs[(size_t)(bT * 18 + 9 + t) * 4096 + hw];
      float px = ox + (float)(t / 3 - 1) + (float)(h + 1);
      float py = oy + (float)(t % 3 - 1) + (float)(w + 1);
      float fx = floorf(px), fy = floorf(py);
      int ltx = min(max((int)fx, 0), 65);
      int lty = min(max((int)fy, 0), 65);
      int rbx = min(max((int)fx + 1, 0), 65);
      int rby = min(max((int)fy + 1, 0), 65);
      float pxc = fminf(fmaxf(px, 0.f), 65.f);
      float pyc = fminf(fmaxf(py, 0.f), 65.f);
      float gx_l = 1.f + ((float)ltx - pxc);
      float gx_r = 1.f - ((float)rbx - pxc);
      float gy_l = 1.f + ((float)lty - pyc);
      float gy_r = 1.f - ((float)rby - pyc);
      int cx[4] = {ltx, rbx, ltx, rbx};
      int cy[4] = {lty, rby, rby, lty};
      float gw[4] = {gx_l * gy_l, gx_r * gy_r, gx_l * gy_r, gx_r * gy_l};
#pragma unroll
      for (int j = 0; j < 4; ++j) {
        bool valid = cx[j] >= 1 && cx[j] <= 64 && cy[j] >= 1 && cy[j] <= 64;
        sidx[j][p] = valid ? ((cx[j] - 1) * 64 + (cy[j] - 1)) : -1;
        swgt[j][p] = gw[j];
      }
    }
    __syncthreads();

    for (int c0 = 0; c0 < 256; c0 += 32) {
      // ---- sample A chunk (32 pixels x 32 channels) into LDS ----
#pragma unroll
      for (int kk0 = 0; kk0 < 32; kk0 += 8) {
        int kk = kk0 + skk;
        const float* xp = x + ((size_t)(bT * 256 + c0 + kk) << 12);
        float v = 0.f;
#pragma unroll
        for (int j = 0; j < 4; ++j) {
          int idx = sidx[j][sp];
          if (idx >= 0) v = fmaf(swgt[j][sp], xp[idx], v);
        }
        AsF[(kk >> 1) * 96 + sp * 2 + (kk & 1)] = v;
      }
      // ---- stage B chunk (32 K x 256 N) from transposed weights ----
      {
        const float* row = wT + ((size_t)((t << 8) + c0 + bkk) << 8);
        float* dst = BsF + (bkk >> 1) * 544 + (bkk & 1);
#pragma unroll
        for (int j = 0; j < 8; ++j) {
          const float4 v = *(const float4*)(row + bnb + (j << 2));
          int n = (bnb + (j << 2)) << 1;
          dst[n] = v.x;
          dst[n + 2] = v.y;
          dst[n + 4] = v.z;
          dst[n + 6] = v.w;
        }
      }
      __syncthreads();
      // ---- WMMA: 8 K-steps of 4, 2x2 tiles per wave ----
#pragma unroll
      for (int k4 = 0; k4 < 32; k4 += 4) {
        int q = (k4 >> 1) + ((lane >> 4) & 1);
        v2f a0 = As2[q * 48 + (lane & 15)];
        v2f a1 = As2[q * 48 + 16 + (lane & 15)];
        v2f b0 = Bs2[q * 272 + n0 + (lane & 15)];
        v2f b1 = Bs2[q * 272 + n0 + 16 + (lane & 15)];
        acc[0][0] = wmma4(a0, b0, acc[0][0]);
        acc[0][1] = wmma4(a0, b1, acc[0][1]);
        acc[1][0] = wmma4(a1, b0, acc[1][0]);
        acc[1][1] = wmma4(a1, b1, acc[1][1]);
      }
      __syncthreads();
    }
  }
  // ---- epilogue: write (m, colOff+n) ----
#pragma unroll
  for (int mi = 0; mi < 2; ++mi)
#pragma unroll
    for (int ni = 0; ni < 2; ++ni) {
      int n = colOff + n0 + ni * 16 + (lane & 15);
      int mb = m0 + mi * 16 + ((lane >> 4) & 1) * 8;
#pragma unroll
      for (int r = 0; r < 8; ++r)
        convout[(size_t)(mb + r) * 512 + n] = acc[mi][ni][r];
    }
}

// ---------------- final 1x1 conv (K=512) + bias + residual ----------------
__global__ __launch_bounds__(256) void k_final_gemm(
    const float* __restrict__ convout,  // (16384,512)
    const float* __restrict__ wTo,      // (512,256) [k][n]
    const float* __restrict__ bo, const float* __restrict__ x,
    float* __restrict__ out) {
  __shared__ v2f As2[16 * 48];
  __shared__ v2f Bs2[16 * 272];
  int tid = threadIdx.x, lane = tid & 31, wv = tid >> 5;
  int m0 = blockIdx.x << 5, bT = m0 >> 12, n0 = wv << 5;
  float* BsF = (float*)Bs2;

  v8f acc[2][2];
#pragma unroll
  for (int mi = 0; mi < 2; ++mi)
#pragma unroll
    for (int ni = 0; ni < 2; ++ni)
#pragma unroll
      for (int r = 0; r < 8; ++r) acc[mi][ni][r] = 0.f;

  int ap = tid >> 3, akb = (tid & 7) << 2;
  int bkk = tid >> 3, bnb = (tid & 7) << 5;

  for (int c0 = 0; c0 < 512; c0 += 32) {
    // A stage: 32 pixels x 32 K, contiguous rows of convout
    {
      const float4 v =
          *(const float4*)(convout + (size_t)(m0 + ap) * 512 + c0 + akb);
      int q = akb >> 1;
      v2f t0;
      t0.x = v.x;
      t0.y = v.y;
      As2[q * 48 + ap] = t0;
      v2f t1;
      t1.x = v.z;
      t1.y = v.w;
      As2[(q + 1) * 48 + ap] = t1;
    }
    // B stage
    {
      const float* row = wTo + ((size_t)(c0 + bkk) << 8);
      float* dst = BsF + (bkk >> 1) * 544 + (bkk & 1);
#pragma unroll
      for (int j = 0; j < 8; ++j) {
        const float4 v = *(const float4*)(row + bnb + (j << 2));
        int n = (bnb + (j << 2)) << 1;
        dst[n] = v.x;
        dst[n + 2] = v.y;
        dst[n + 4] = v.z;
        dst[n + 6] = v.w;
      }
    }
    __syncthreads();
#pragma unroll
    for (int k4 = 0; k4 < 32; k4 += 4) {
      int q = (k4 >> 1) + ((lane >> 4) & 1);
      v2f a0 = As2[q * 48 + (lane & 15)];
      v2f a1 = As2[q * 48 + 16 + (lane & 15)];
      v2f b0 = Bs2[q * 272 + n0 + (lane & 15)];
      v2f b1 = Bs2[q * 272 + n0 + 16 + (lane & 15)];
      acc[0][0] = wmma4(a0, b0, acc[0][0]);
      acc[0][1] = wmma4(a0, b1, acc[0][1]);
      acc[1][0] = wmma4(a1, b0, acc[1][0]);
      acc[1][1] = wmma4(a1, b1, acc[1][1]);
    }
    __syncthreads();
  }
  // epilogue: bias + residual, NCHW output
#pragma unroll
  for (int mi = 0; mi < 2; ++mi)
#pragma unroll
    for (int ni = 0; ni < 2; ++ni) {
      int n = n0 + ni * 16 + (lane & 15);
      float bias = bo[n];
      int mb = m0 + mi * 16 + ((lane >> 4) & 1) * 8;
#pragma unroll
      for (int r = 0; r < 8; ++r) {
        int m = mb + r;
        int hw = m & 4095;
        size_t oi = ((size_t)(bT * 256 + n) << 12) + hw;
        out[oi] = acc[mi][ni][r] + bias + x[oi];
      }
    }
}

// ---------------- host-side launcher ----------------
extern "C" void kernel_launch(void* const* d_in, const int* in_sizes, int n_in,
                              void* d_out, int out_size, void* d_ws,
                              size_t ws_size, hipStream_t stream) {
  (void)in_sizes;
  (void)n_in;
  (void)out_size;
  (void)ws_size;
  const float* x   = (const float*)d_in[0];
  const float* pwx = (const float*)d_in[1];
  const float* pbx = (const float*)d_in[2];
  const float* wx  = (const float*)d_in[3];
  const float* pwy = (const float*)d_in[4];
  const float* pby = (const float*)d_in[5];
  const float* wy  = (const float*)d_in[6];
  const float* wo  = (const float*)d_in[7];
  const float* bo  = (const float*)d_in[8];
  float* out = (float*)d_out;
  float* ws = (float*)d_ws;

  float* wT_x = ws + OFF_WT_X;
  float* wT_y = ws + OFF_WT_Y;
  float* wT_o = ws + OFF_WT_O;
  float* pwT  = ws + OFF_PWT;
  float* offs = ws + OFF_OFFS;  // branch 0 then branch 1 (stride 294912)
  float* conv = ws + OFF_CONV;  // (16384, 512)

  k_transpose_main_w<<<(9 * 256 * 256 + 255) / 256, 256, 0, stream>>>(wx, wT_x);
  k_transpose_main_w<<<(9 * 256 * 256 + 255) / 256, 256, 0, stream>>>(wy, wT_y);
  k_transpose_wo<<<(512 * 256 + 255) / 256, 256, 0, stream>>>(wo, wT_o);
  k_build_pwt<<<(256 * 36 * 9 + 255) / 256, 256, 0, stream>>>(pwx, pwy, pwT);

  k_offset_conv<<<256, 256, 0, stream>>>(x, pwT, pbx, pby, offs);

  k_deform_gemm<<<512, 256, 0, stream>>>(x, offs, wT_x, conv, 0);
  k_deform_gemm<<<512, 256, 0, stream>>>(x, offs + (size_t)4 * 18 * 4096, wT_y,
                                         conv, 256);

  k_final_gemm<<<512, 256, 0, stream>>>(conv, wT_o, bo, x, out);
}